// GinEncoder_80882824118337
// MI455X (gfx1250) — compile-verified
//
#include <hip/hip_runtime.h>
#include <hip/hip_bf16.h>

typedef __attribute__((ext_vector_type(2))) float v2f;
typedef __attribute__((ext_vector_type(8))) float v8f;

#define NNODES 100000
#define NEDGES 1600000
#define FIN    128
#define HID    64
#define NGRAPH 100
#define NLAYER 3
#define BN_EPS 1e-5f

// ---------------- utility kernels ----------------

__global__ void zero_f32(float* __restrict__ p, int n) {
    int i = blockIdx.x * blockDim.x + threadIdx.x;
    if (i < n) p[i] = 0.0f;
}

__global__ void copy_f32v4(float* __restrict__ dst, const float* __restrict__ src, int n4) {
    int i = blockIdx.x * blockDim.x + threadIdx.x;
    if (i < n4) ((float4*)dst)[i] = ((const float4*)src)[i];
}

// ---------------- edge aggregation: z[dst] += h[src] ----------------
// One thread handles one (edge, 4-feature chunk). h array is L2-resident
// (<= 51 MB vs 192 MB L2) so gathers hit L2; scatter uses f32 L2 atomics.

__global__ void edge_agg(const float* __restrict__ h, const int* __restrict__ ei,
                         float* __restrict__ z, int shift) {
    long long idx = (long long)blockIdx.x * blockDim.x + threadIdx.x;
    int e   = (int)(idx >> shift);
    int c   = ((int)idx & ((1 << shift) - 1)) * 4;
    int din = 4 << shift;
    int src = ei[e];
    int dst = ei[NEDGES + e];
    const float4 v = *(const float4*)(h + (long long)src * din + c);
    float* o = z + (long long)dst * din + c;
    atomicAdd(o + 0, v.x);
    atomicAdd(o + 1, v.y);
    atomicAdd(o + 2, v.z);
    atomicAdd(o + 3, v.w);
}

// ---------------- fp32 WMMA GEMM: Y = Z @ W + b, plus column stats ----------------
// One wave32 computes a 16x16 output tile with V_WMMA_F32_16X16X4_F32.
// A (16x4): lanes 0-15 hold row M=lane, K={k,k+1}; lanes 16-31 hold K={k+2,k+3}.
// B (4x16): lanes 0-15 hold col N=lane, K={k,k+1}; lanes 16-31 hold K={k+2,k+3}.
// C/D (16x16): lane = column N (mod 16), VGPR v = row M (v, or v+8 for hi lanes).

__global__ void __launch_bounds__(256) gemm_wmma_f32(
    const float* __restrict__ Z, const float* __restrict__ W,
    const float* __restrict__ bias, float* __restrict__ Y,
    float* __restrict__ colsum, float* __restrict__ colsq, int din) {
    int lane  = threadIdx.x & 31;
    int wave  = blockIdx.x * 8 + (threadIdx.x >> 5);
    int tileM = (wave >> 2) << 4;       // 6250 row tiles (N divisible by 16)
    int tileN = (wave & 3) << 4;        // 4 col tiles of 16 (H = 64)
    int half  = lane >> 4;              // 0: K pair {0,1}, 1: K pair {2,3}
    int l     = lane & 15;

    const float* arow = Z + (long long)(tileM + l) * din + half * 2;   // 8B aligned
    const float* bcol = W + (long long)(half * 2) * HID + (tileN + l);

    v8f c = {};
    for (int k = 0; k < din; k += 4) {
        v2f a = *(const v2f*)(arow + k);                 // Z[row][k+2h .. k+2h+1]
        v2f b = { bcol[k * HID], bcol[k * HID + HID] };  // W[k+2h][col], W[k+2h+1][col]
        c = __builtin_amdgcn_wmma_f32_16x16x4_f32(
                /*neg_a=*/false, a, /*neg_b=*/false, b,
                /*c_mod=*/(short)0, c, /*reuse_a=*/false, /*reuse_b=*/false);
    }

    int   col = tileN + l;
    float bv  = bias[col];
    float s = 0.0f, sq = 0.0f;
    float* yo = Y + (long long)(tileM + half * 8) * HID + col;
#pragma unroll
    for (int v = 0; v < 8; ++v) {
        float t = c[v] + bv;
        s  += t;
        sq += t * t;
        yo[(long long)v * HID] = t;
    }
    // each lane owns a single column -> one atomic per stat per lane
    atomicAdd(&colsum[col], s);
    atomicAdd(&colsq[col],  sq);
}

// ---------------- BN stat finalize (64 threads) ----------------

__global__ void bn_finalize(const float* __restrict__ sum, const float* __restrict__ sq,
                            float* __restrict__ mean, float* __restrict__ rstd) {
    int c = threadIdx.x;
    float m = sum[c] * (1.0f / NNODES);
    float v = sq[c] * (1.0f / NNODES) - m * m;   // biased variance (PyTorch BN)
    mean[c] = m;
    rstd[c] = rsqrtf(v + BN_EPS);
}

// ---------------- BN + ReLU (+ optional global_add_pool) ----------------

__global__ void bn_relu_pool(const float* __restrict__ Y,
                             const float* __restrict__ mean, const float* __restrict__ rstd,
                             const float* __restrict__ gamma, const float* __restrict__ beta,
                             const int* __restrict__ batch,
                             float* __restrict__ hout, float* __restrict__ out,
                             int layerOff, int doPool) {
    int idx = blockIdx.x * blockDim.x + threadIdx.x;  // N*HID threads
    int n = idx >> 6;
    int c = idx & 63;
    float v = (Y[idx] - mean[c]) * rstd[c] * gamma[c] + beta[c];
    v = fmaxf(v, 0.0f);
    hout[idx] = v;
    if (doPool) {
        int g = batch[n];
        atomicAdd(&out[g * (NLAYER * HID) + layerOff + c], v);
    }
}

// ---------------- driver ----------------

extern "C" void kernel_launch(void* const* d_in, const int* in_sizes, int n_in,
                              void* d_out, int out_size, void* d_ws, size_t ws_size,
                              hipStream_t stream) {
    (void)in_sizes; (void)n_in; (void)out_size; (void)ws_size;

    const float* x     = (const float*)d_in[0];
    const int*   ei    = (const int*)d_in[1];
    const int*   batch = (const int*)d_in[2];

    const float *Wp[NLAYER][2], *bp[NLAYER][2], *gp[NLAYER][2], *bep[NLAYER][2];
    int k = 3;
    for (int i = 0; i < NLAYER; ++i)
        for (int j = 0; j < 2; ++j) {
            Wp[i][j]  = (const float*)d_in[k++];
            bp[i][j]  = (const float*)d_in[k++];
            gp[i][j]  = (const float*)d_in[k++];
            bep[i][j] = (const float*)d_in[k++];
        }

    float* ws     = (float*)d_ws;
    float* z      = ws;                              // N*128
    float* y      = z  + (size_t)NNODES * FIN;       // N*64 (pre-BN GEMM out)
    float* hA     = y  + (size_t)NNODES * HID;       // N*64
    float* hB     = hA + (size_t)NNODES * HID;       // N*64
    float* colsum = hB + (size_t)NNODES * HID;       // 64
    float* colsq  = colsum + HID;                    // 64
    float* mean   = colsq  + HID;                    // 64
    float* rstd   = mean   + HID;                    // 64
    float* out    = (float*)d_out;

    zero_f32<<<(NGRAPH * NLAYER * HID + 255) / 256, 256, 0, stream>>>(
        out, NGRAPH * NLAYER * HID);

    const float* hcur = x;
    for (int i = 0; i < NLAYER; ++i) {
        const int din   = (i == 0) ? FIN : HID;
        const int shift = (i == 0) ? 5 : 4;          // din/4 chunks per edge

        // z = hcur ; z[dst] += hcur[src]  (GINConv eps=0, aggr='add')
        const int n4 = NNODES * din / 4;
        copy_f32v4<<<(n4 + 255) / 256, 256, 0, stream>>>(z, hcur, n4);
        const long long tE = (long long)NEDGES << shift;
        edge_agg<<<(int)(tE / 256), 256, 0, stream>>>(hcur, ei, z, shift);

        const float* in0 = z;
        for (int j = 0; j < 2; ++j) {
            const int dj = (j == 0) ? din : HID;
            zero_f32<<<1, 128, 0, stream>>>(colsum, 128);  // colsum + colsq contiguous
            gemm_wmma_f32<<<(NNODES / 16) * (HID / 16) / 8, 256, 0, stream>>>(
                in0, Wp[i][j], bp[i][j], y, colsum, colsq, dj);
            bn_finalize<<<1, HID, 0, stream>>>(colsum, colsq, mean, rstd);
            float* hdst = (j == 0) ? hA : hB;
            bn_relu_pool<<<NNODES * HID / 256, 256, 0, stream>>>(
                y, mean, rstd, gp[i][j], bep[i][j], batch, hdst, out,
                i * HID, (j == 1) ? 1 : 0);
            in0 = hdst;
        }
        hcur = hB;
    }
}